// Solver_22565758173707
// MI455X (gfx1250) — compile-verified
//
#include <hip/hip_runtime.h>

// ---------------------------------------------------------------------------
// Batched 19x19 Newton step (Hadamard S-matrix Poisson solver) for gfx1250.
// fp32 WMMA (V_WMMA_F32_16X16X4_F32) builds the per-pixel Hessian H = A^T A,
// A = diag(u*sqrt(1-eps)) * Sinv; per-lane Cholesky solve replaces linalg.inv.
// Round 3: (1-eps) folded into WMMA operands, diagonal terms added inside the
// Cholesky, and packed scatter indices hoisted out of the pixel loop -> the
// WMMA loop is fully branchless (plain ds_store_b32, no exec masks).
// ---------------------------------------------------------------------------

typedef float v2f __attribute__((ext_vector_type(2)));
typedef float v8f __attribute__((ext_vector_type(8)));

constexpr int N   = 19;   // S-matrix order
constexpr int NP  = 20;   // K padded to multiple of 4 for 16x16x4 WMMA
constexpr int SC  = 33;   // Sinv LDS column stride (covers cols 0..31, odd skew)
constexpr int PPB = 64;   // pixels per block = 2 waves * 32 lanes
constexpr int HS  = 191;  // packed lower-tri stride (0..189 used, 190 = dump slot)

__launch_bounds__(PPB)
__global__ void newton19_kernel(const float* __restrict__ rate,
                                const float* __restrict__ pixels,
                                const float* __restrict__ SinvG,
                                const float* __restrict__ tP,
                                const float* __restrict__ ssP,
                                const float* __restrict__ epsP,
                                float* __restrict__ outRate,
                                float* __restrict__ outRes,
                                int nPix)
{
    __shared__ float sinv[NP][SC];     // zero-padded Sinv (row 19 and cols 19..32 are 0)
    __shared__ float sbuf[PPB][NP];    // u*sqrt(1-eps) per pixel (col 19 = 0 pad)
    __shared__ float dbuf[PPB][NP];    // (1-eps)*diag_vals + eps per pixel
    __shared__ float hbuf[PPB][HS];    // packed lower-tri H -> Cholesky L (diag holds 1/Ljj)

    const int tid  = threadIdx.x;
    const int lane = tid & 31;
    const int wave = tid >> 5;
    const int m16  = lane & 15;
    const int hi   = lane >> 4;                 // half-wave select (K-pair / row+8)

    const int  p     = blockIdx.x * PPB + tid;  // this lane's pixel
    const bool valid = (p < nPix);
    const int  pc    = valid ? p : (nPix - 1);  // clamped for loads

    const float t    = tP[0];
    const float ss   = ssP[0];
    const float eps  = epsP[0];
    const float ome  = 1.0f - eps;
    const float some = sqrtf(ome);              // sqrt(1-eps): squared by the WMMA

    // ---- stage Sinv into LDS with zero padding --------------------------------
    for (int i = tid; i < NP * SC; i += PPB) {
        int rI = i / SC, cI = i % SC;
        sinv[rI][cI] = (rI < N && cI < N) ? SinvG[rI * N + cI] : 0.0f;
    }
    __syncthreads();

    // ---- per-lane scalar phase: x = Sinv r, u, J, diag ------------------------
    float r[N], k[N];
#pragma unroll
    for (int j = 0; j < N; ++j) { r[j] = rate[pc * N + j]; k[j] = pixels[pc * N + j]; }

    float u[N];
#pragma unroll
    for (int a = 0; a < N; ++a) {
        float acc = 0.0f;
#pragma unroll
        for (int b = 0; b < N; ++b) acc = fmaf(sinv[a][b], r[b], acc);
        u[a] = 1.0f / (acc + eps);              // 1/(x+eps); w = u^2
    }

    float J[N];
#pragma unroll
    for (int b = 0; b < N; ++b) {
        float g = 0.0f;
#pragma unroll
        for (int a = 0; a < N; ++a) g = fmaf(sinv[a][b], u[a], g);   // (Sinv^T u)[b]
        const float rb = r[b];
        J[b] = t * (1.0f - k[b] / (rb + eps)) - g - 1.0f / (rb + eps);
        // pre-scaled Hessian diagonal term: (1-eps)*diag_vals[b] + eps
        dbuf[tid][b] = ome * ((t * k[b] + 1.0f) / (rb * rb + eps)) + eps;
        sbuf[tid][b] = u[b] * some;             // folds (1-eps) into A^T A
    }
    sbuf[tid][N] = 0.0f;                        // K=19 padding
    dbuf[tid][N] = 0.0f;
    __syncthreads();

    // ---- precompute packed lower-tri scatter indices (pixel-invariant) --------
    // C/D tile layout: row = bt*16 + v + 8*hi, col = ct*16 + m16. Entries outside
    // the 19x19 lower triangle go to the dump slot HS-1 (never read).
    int idx00[8], idx10[8], idx11[8];
#pragma unroll
    for (int v = 0; v < 8; ++v) {
        const int rv = v + 8 * hi;
        {   // tile (0,0): rows 0-15, cols 0-15
            const int row = rv, col = m16;
            idx00[v] = ((col <= row) ? (row * (row + 1) / 2 + col) : (HS - 1));
        }
        {   // tile (1,0): rows 16-31, cols 0-15
            const int row = 16 + rv, col = m16;
            idx10[v] = ((row < N) ? (row * (row + 1) / 2 + col) : (HS - 1));
        }
        {   // tile (1,1): rows 16-31, cols 16-31
            const int row = 16 + rv, col = 16 + m16;
            idx11[v] = ((row < N && col <= row) ? (row * (row + 1) / 2 + col) : (HS - 1));
        }
    }

    // ---- WMMA phase: per pixel (1-eps)*A^T A ----------------------------------
    // With the f32 16x16x4 operand stripings, the A-operand of row-tile T equals
    // the B-operand of col-tile T, so only two operand vectors per k-step.
    for (int px = wave * 32; px < wave * 32 + 32; ++px) {
        v8f a00 = {0.f,0.f,0.f,0.f,0.f,0.f,0.f,0.f};
        v8f a10 = {0.f,0.f,0.f,0.f,0.f,0.f,0.f,0.f};
        v8f a11 = {0.f,0.f,0.f,0.f,0.f,0.f,0.f,0.f};
#pragma unroll
        for (int k0 = 0; k0 < NP; k0 += 4) {
            const int   kb = k0 + 2 * hi;                 // this half-wave's K pair
            const float ua = sbuf[px][kb];
            const float ub = sbuf[px][kb + 1];
            v2f P0, P1;                                   // cols 0-15 / 16-31 of A
            P0.x = ua * sinv[kb][m16];
            P0.y = ub * sinv[kb + 1][m16];
            P1.x = ua * sinv[kb][16 + m16];
            P1.y = ub * sinv[kb + 1][16 + m16];
            a00 = __builtin_amdgcn_wmma_f32_16x16x4_f32(false, P0, false, P0, (short)0, a00, false, false);
            a10 = __builtin_amdgcn_wmma_f32_16x16x4_f32(false, P1, false, P0, (short)0, a10, false, false);
            a11 = __builtin_amdgcn_wmma_f32_16x16x4_f32(false, P1, false, P1, (short)0, a11, false, false);
        }
        // Fully branchless scatter: raw accumulator values to precomputed slots.
        float* __restrict__ hrow = &hbuf[px][0];
#pragma unroll
        for (int v = 0; v < 8; ++v) {
            hrow[idx00[v]] = a00[v];
            hrow[idx10[v]] = a10[v];
            hrow[idx11[v]] = a11[v];
        }
    }
    __syncthreads();

    // ---- per-lane Cholesky (in-place, packed; diag stores 1/Ljj) --------------
    // Diagonal terms (1-eps)*diag_vals + eps are added here, off the WMMA path.
    float* __restrict__ Hp = &hbuf[tid][0];
    for (int j = 0; j < N; ++j) {
        const int jj = j * (j + 1) / 2;
        float d = Hp[jj + j] + dbuf[tid][j];
        for (int m = 0; m < j; ++m) { const float L = Hp[jj + m]; d = fmaf(-L, L, d); }
        const float inv = rsqrtf(d);
        Hp[jj + j] = inv;
        for (int i = j + 1; i < N; ++i) {
            const int ii = i * (i + 1) / 2;
            float s = Hp[ii + j];
            for (int m = 0; m < j; ++m) s = fmaf(-Hp[ii + m], Hp[jj + m], s);
            Hp[ii + j] = s * inv;
        }
    }

    // ---- triangular solves (fully unrolled, static indices -> registers) ------
    float y[N];
#pragma unroll
    for (int j = 0; j < N; ++j) {
        float s = J[j];
#pragma unroll
        for (int m = 0; m < j; ++m) s = fmaf(-Hp[j * (j + 1) / 2 + m], y[m], s);
        y[j] = s * Hp[j * (j + 1) / 2 + j];
    }
    float z[N];
#pragma unroll
    for (int i = N - 1; i >= 0; --i) {
        float s = y[i];
#pragma unroll
        for (int m = i + 1; m < N; ++m) s = fmaf(-Hp[m * (m + 1) / 2 + i], z[m], s);
        z[i] = s * Hp[i * (i + 1) / 2 + i];
    }

    float res = 0.0f;
#pragma unroll
    for (int e = 0; e < N; ++e) res = fmaf(J[e], z[e], res);   // J^T Hinv J

    if (valid) {
#pragma unroll
        for (int j = 0; j < N; ++j)
            outRate[p * N + j] = r[j] - ss * z[j];             // rate + step_size*(-Hinv J)
        outRes[p] = res;
    }
}

extern "C" void kernel_launch(void* const* d_in, const int* in_sizes, int n_in,
                              void* d_out, int out_size, void* d_ws, size_t ws_size,
                              hipStream_t stream)
{
    const float* rate = (const float*)d_in[0];
    const float* pix  = (const float*)d_in[1];
    const float* Sinv = (const float*)d_in[2];
    const float* t    = (const float*)d_in[3];
    const float* ss   = (const float*)d_in[4];
    const float* eps  = (const float*)d_in[5];

    const int nRateElems = in_sizes[0];        // H*W*N
    const int nPix       = nRateElems / N;     // H*W

    float* outRate = (float*)d_out;            // first output: rate_new (flat)
    float* outRes  = outRate + nRateElems;     // second output: residual (flat)

    const int blocks = (nPix + PPB - 1) / PPB;
    newton19_kernel<<<blocks, PPB, 0, stream>>>(rate, pix, Sinv, t, ss, eps,
                                                outRate, outRes, nPix);
}